// PredictYolov1_2542620639984
// MI455X (gfx1250) — compile-verified
//
#include <hip/hip_runtime.h>
#include <hip/hip_bf16.h>

typedef float v2f __attribute__((ext_vector_type(2)));
typedef float v8f __attribute__((ext_vector_type(8)));

#define BB 1024
#define GG 28
#define NBX 2
#define NCL 20
#define KDET 4096
#define NTOT (BB * GG * GG * NBX)   // 1,605,632
#define THR_CONF 3.0f
#define THR_NMS 0.3f
#define NEGV -1e30f
#define MASK_WORDS_PER_ROW 128      // 4096 bits / 32
#define MASK_TOTAL_WORDS (KDET * MASK_WORDS_PER_ROW)

struct Ctl {
    unsigned long long prefix;
    unsigned int kth;
    unsigned int counter;
    unsigned int pad;
};

// ---------------------------------------------------------------- keys
__global__ void k_keys(const float* __restrict__ confs,
                       unsigned long long* __restrict__ keys) {
    int i = blockIdx.x * blockDim.x + threadIdx.x;
    if (i >= NTOT) return;
    float c = confs[i];
    float m = (c > THR_CONF) ? c : NEGV;
    unsigned int u = __float_as_uint(m);
    u ^= (u & 0x80000000u) ? 0xFFFFFFFFu : 0x80000000u;   // monotonic key
    keys[i] = ((unsigned long long)u << 32) | (unsigned int)(~(unsigned int)i);
}

__global__ void k_init(Ctl* ctl) {
    if (threadIdx.x == 0) { ctl->prefix = 0ull; ctl->kth = KDET; ctl->counter = 0u; }
}

__global__ void k_zero_hist(unsigned int* hist) { hist[threadIdx.x] = 0u; }

// ---------------------------------------------------------------- radix select
__global__ void k_hist(const unsigned long long* __restrict__ keys,
                       const Ctl* __restrict__ ctl,
                       unsigned int* __restrict__ hist,
                       int shift, unsigned long long maskHigh) {
    __shared__ unsigned int lh[256];
    int tid = threadIdx.x;
    lh[tid] = 0u;
    __syncthreads();
    unsigned long long prefix = ctl->prefix;
    int i = blockIdx.x * blockDim.x + tid;
    if (i < NTOT) {
        unsigned long long k = keys[i];
        if (((k ^ prefix) & maskHigh) == 0ull) {
            atomicAdd(&lh[(unsigned int)((k >> shift) & 255ull)], 1u);
        }
    }
    __syncthreads();
    if (lh[tid]) atomicAdd(&hist[tid], lh[tid]);
}

__global__ void k_select(const unsigned int* __restrict__ hist, Ctl* ctl, int shift) {
    if (threadIdx.x != 0) return;
    unsigned int kth = ctl->kth;
    unsigned int cum = 0;
    for (int d = 255; d >= 0; --d) {
        unsigned int h = hist[d];
        if (cum + h >= kth) {
            ctl->prefix |= ((unsigned long long)(unsigned int)d) << shift;
            ctl->kth = kth - cum;
            break;
        }
        cum += h;
    }
    ctl->counter = 0u;
}

__global__ void k_compact(const unsigned long long* __restrict__ keys,
                          Ctl* ctl, unsigned long long* __restrict__ cand) {
    int i = blockIdx.x * blockDim.x + threadIdx.x;
    if (i >= NTOT) return;
    unsigned long long k = keys[i];
    if (k >= ctl->prefix) {
        unsigned int pos = atomicAdd(&ctl->counter, 1u);
        if (pos < KDET) cand[pos] = k;     // exactly KDET land here (keys unique)
    }
}

// ---------------------------------------------------------------- bitonic sort (desc)
__global__ __launch_bounds__(1024) void k_sort(unsigned long long* cand) {
    __shared__ unsigned long long s[KDET];
    int tid = threadIdx.x;
    for (int t = tid; t < KDET; t += 1024) s[t] = cand[t];
    __syncthreads();
    for (int k = 2; k <= KDET; k <<= 1) {
        for (int j = k >> 1; j > 0; j >>= 1) {
            for (int t = tid; t < KDET; t += 1024) {
                int x = t ^ j;
                if (x > t) {
                    bool desc = (t & k) == 0;
                    unsigned long long u = s[t], v = s[x];
                    if (desc ? (u < v) : (u > v)) { s[t] = v; s[x] = u; }
                }
            }
            __syncthreads();
        }
    }
    for (int t = tid; t < KDET; t += 1024) cand[t] = s[t];
}

// ---------------------------------------------------------------- gather + decode
__global__ void k_decode(const unsigned long long* __restrict__ cand,
                         const float* __restrict__ p_boxes,
                         const float* __restrict__ p_confs,
                         const float* __restrict__ p_clses,
                         float* __restrict__ out,
                         float4* __restrict__ boxes4,
                         float* __restrict__ area,
                         int* __restrict__ group,
                         int* __restrict__ validArr) {
    int r = blockIdx.x * blockDim.x + threadIdx.x;
    if (r >= KDET) return;
    unsigned long long key = cand[r];
    unsigned int idx = ~((unsigned int)(key & 0xFFFFFFFFull));
    float conf = p_confs[idx];
    int valid = (conf > THR_CONF) ? 1 : 0;

    int nb = idx & (NBX - 1);
    int cell = idx >> 1;                 // (b*G+gy)*G + gx
    int gx = cell % GG;
    int t1 = cell / GG;
    int gy = t1 % GG;
    int b  = t1 / GG;

    const float* bp = p_boxes + (size_t)cell * (NBX * 4) + nb * 4;
    float cx = (bp[0] + (float)gx) / (float)GG;
    float cy = (bp[1] + (float)gy) / (float)GG;
    float w = bp[2], h = bp[3];
    float l  = cx - w * 0.5f;
    float tt = cy - h * 0.5f;
    float rr = cx + w * 0.5f;
    float bt = cy + h * 0.5f;

    const float* cp = p_clses + (size_t)cell * NCL;
    float best = cp[0]; int bestc = 0;
    #pragma unroll
    for (int c = 1; c < NCL; ++c) {
        float v = cp[c];
        if (v > best) { best = v; bestc = c; }
    }
    int label = bestc + 1;

    // outputs: bids | boxes | labels | scores | keep
    out[r] = (float)b;
    out[KDET + 4 * r + 0] = l;
    out[KDET + 4 * r + 1] = tt;
    out[KDET + 4 * r + 2] = rr;
    out[KDET + 4 * r + 3] = bt;
    out[KDET * 5 + r] = (float)label;
    out[KDET * 6 + r] = conf;

    boxes4[r] = make_float4(l, tt, rr, bt);
    area[r] = fmaxf(rr - l, 0.0f) * fmaxf(bt - tt, 0.0f);
    group[r] = b * (NCL + 1) + label;
    validArr[r] = valid;
}

__global__ void k_zero_mask(unsigned int* mask) {
    int i = blockIdx.x * blockDim.x + threadIdx.x;
    if (i < MASK_TOTAL_WORDS) mask[i] = 0u;
}

// ---------------------------------------------------------------- suppress tiles (WMMA)
// Each wave32 handles one 16x16 tile. Pairwise area sums (union term) come from the
// matrix unit: D = A(16x4) * B(4x16), A=[area_row,1,0,0], B=[1;area_col;0;0], so
// D[m][n] = area_row[m] + area_col[n] exactly (products are x1.0, f32 RNE).
__global__ __launch_bounds__(256) void k_tiles(const float* __restrict__ area,
                                               const float4* __restrict__ boxes,
                                               const int* __restrict__ group,
                                               unsigned int* __restrict__ mask) {
    int wave = blockIdx.x * 8 + (threadIdx.x >> 5);
    int lane = threadIdx.x & 31;
    int ti = wave >> 8;          // 256 tiles per dim
    int tj = wave & 255;
    int half = lane >> 4;
    int ln = lane & 15;
    int rbase = ti * 16, cbase = tj * 16;

    // A layout (16x4 f32): lanes 0-15 hold K=0 (v0), K=1 (v1); lanes 16-31 hold K=2,3 (zeros)
    v2f a, b;
    a.x = half ? 0.0f : area[rbase + ln];
    a.y = half ? 0.0f : 1.0f;
    // B layout (4x16 f32): lanes 0-15 hold K=0 (v0), K=1 (v1); lanes 16-31 hold K=2,3 (zeros)
    b.x = half ? 0.0f : 1.0f;
    b.y = half ? 0.0f : area[cbase + ln];
    v8f c = {0.f, 0.f, 0.f, 0.f, 0.f, 0.f, 0.f, 0.f};
    c = __builtin_amdgcn_wmma_f32_16x16x4_f32(false, a, false, b, (short)0, c, false, false);
    // constant-index extraction (keeps one static WMMA, no dynamic vector indexing)
    float c0 = c[0], c1 = c[1], c2 = c[2], c3 = c[3];
    float c4 = c[4], c5 = c[5], c6 = c[6], c7 = c[7];

    int n = cbase + ln;
    float4 bc = boxes[n];
    int gc = group[n];
    int wc = tj >> 1;
    int sh = (tj & 1) * 16;

#define TILE_BODY(V, CV)                                                              \
    {                                                                                 \
        int m = rbase + (V) + 8 * half;                                               \
        float4 br = boxes[m];                                                         \
        float ltx = fmaxf(br.x, bc.x);                                                \
        float lty = fmaxf(br.y, bc.y);                                                \
        float rbx = fminf(br.z, bc.z);                                                \
        float rby = fminf(br.w, bc.w);                                                \
        float inter = fmaxf(rbx - ltx, 0.0f) * fmaxf(rby - lty, 0.0f);                \
        float uni = fmaxf((CV) - inter, 1e-9f);                                       \
        float iou = inter / uni;                                                      \
        int pred = (group[m] == gc) && (iou > THR_NMS);                               \
        unsigned long long bal = __ballot(pred);                                      \
        if (lane == 0) {                                                              \
            unsigned int lo = (unsigned int)(bal & 0xFFFFull);                        \
            unsigned int hi = (unsigned int)((bal >> 16) & 0xFFFFull);                \
            if (lo) atomicOr(&mask[(size_t)(rbase + (V)) * MASK_WORDS_PER_ROW + wc],  \
                             lo << sh);                                               \
            if (hi) atomicOr(&mask[(size_t)(rbase + (V) + 8) * MASK_WORDS_PER_ROW + wc], \
                             hi << sh);                                               \
        }                                                                             \
    }

    TILE_BODY(0, c0)
    TILE_BODY(1, c1)
    TILE_BODY(2, c2)
    TILE_BODY(3, c3)
    TILE_BODY(4, c4)
    TILE_BODY(5, c5)
    TILE_BODY(6, c6)
    TILE_BODY(7, c7)
#undef TILE_BODY
}

// ---------------------------------------------------------------- serial NMS (one wave32)
__global__ __launch_bounds__(32) void k_nms(const unsigned int* __restrict__ mask,
                                            const int* __restrict__ validArr,
                                            float* __restrict__ out) {
    int l = threadIdx.x;   // lane 0..31; lane owns words 4l..4l+3
    // build valid bitmask fragments
    unsigned int vw0 = 0, vw1 = 0, vw2 = 0, vw3 = 0;
    int base = (4 * l) * 32;
    for (int k = 0; k < 32; ++k) {
        vw0 |= (validArr[base + k]        ? 1u : 0u) << k;
        vw1 |= (validArr[base + 32 + k]   ? 1u : 0u) << k;
        vw2 |= (validArr[base + 64 + k]   ? 1u : 0u) << k;
        vw3 |= (validArr[base + 96 + k]   ? 1u : 0u) << k;
    }
    unsigned int kw0 = 0, kw1 = 0, kw2 = 0, kw3 = 0;

    const uint4* mrows = reinterpret_cast<const uint4*>(mask);
    uint4 cur = mrows[(size_t)0 * 32 + l];
    for (int i = 0; i < KDET; ++i) {
        uint4 nxt = (i + 1 < KDET) ? mrows[(size_t)(i + 1) * 32 + l]
                                   : make_uint4(0u, 0u, 0u, 0u);
        if (i + 8 < KDET) __builtin_prefetch(&mrows[(size_t)(i + 8) * 32 + l], 0, 0);
        // kept bits only exist for j<i, so (idx<i) mask is implicit
        unsigned int w = (cur.x & kw0) | (cur.y & kw1) | (cur.z & kw2) | (cur.w & kw3);
        int any = __any(w != 0u);
        int w0 = i >> 5;
        if (l == (w0 >> 2)) {
            int cc = w0 & 3;
            unsigned int bit = 1u << (i & 31);
            if (!any) {
                switch (cc) {
                    case 0: if (vw0 & bit) kw0 |= bit; break;
                    case 1: if (vw1 & bit) kw1 |= bit; break;
                    case 2: if (vw2 & bit) kw2 |= bit; break;
                    case 3: if (vw3 & bit) kw3 |= bit; break;
                }
            }
        }
        cur = nxt;
    }

    float* keep_out = out + KDET * 7;
    for (int k = 0; k < 32; ++k) {
        keep_out[base + k]      = (kw0 >> k) & 1u ? 1.0f : 0.0f;
        keep_out[base + 32 + k] = (kw1 >> k) & 1u ? 1.0f : 0.0f;
        keep_out[base + 64 + k] = (kw2 >> k) & 1u ? 1.0f : 0.0f;
        keep_out[base + 96 + k] = (kw3 >> k) & 1u ? 1.0f : 0.0f;
    }
}

// ---------------------------------------------------------------- launch
extern "C" void kernel_launch(void* const* d_in, const int* in_sizes, int n_in,
                              void* d_out, int out_size, void* d_ws, size_t ws_size,
                              hipStream_t stream) {
    const float* p_boxes = (const float*)d_in[0];
    const float* p_confs = (const float*)d_in[1];
    const float* p_clses = (const float*)d_in[2];
    float* out = (float*)d_out;

    char* ws = (char*)d_ws;
    size_t off = 0;
    auto carve = [&](size_t bytes) -> char* {
        char* p = ws + off;
        off = (off + bytes + 255) & ~(size_t)255;
        return p;
    };
    unsigned long long* keys = (unsigned long long*)carve((size_t)NTOT * 8);
    unsigned int* hist       = (unsigned int*)carve(256 * 4);
    Ctl* ctl                 = (Ctl*)carve(sizeof(Ctl));
    unsigned long long* cand = (unsigned long long*)carve((size_t)KDET * 8);
    float4* boxes4           = (float4*)carve((size_t)KDET * 16);
    float* area              = (float*)carve((size_t)KDET * 4);
    int* group               = (int*)carve((size_t)KDET * 4);
    int* validArr            = (int*)carve((size_t)KDET * 4);
    unsigned int* mask       = (unsigned int*)carve((size_t)MASK_TOTAL_WORDS * 4);
    (void)ws_size; (void)in_sizes; (void)n_in; (void)out_size;

    const int NBLK = (NTOT + 255) / 256;   // 6272

    k_keys<<<NBLK, 256, 0, stream>>>(p_confs, keys);
    k_init<<<1, 1, 0, stream>>>(ctl);
    for (int pass = 0; pass < 8; ++pass) {
        int shift = 56 - pass * 8;
        unsigned long long maskHigh = (pass == 0) ? 0ull : (~0ull) << (shift + 8);
        k_zero_hist<<<1, 256, 0, stream>>>(hist);
        k_hist<<<NBLK, 256, 0, stream>>>(keys, ctl, hist, shift, maskHigh);
        k_select<<<1, 1, 0, stream>>>(hist, ctl, shift);
    }
    k_compact<<<NBLK, 256, 0, stream>>>(keys, ctl, cand);
    k_sort<<<1, 1024, 0, stream>>>(cand);
    k_decode<<<KDET / 256, 256, 0, stream>>>(cand, p_boxes, p_confs, p_clses,
                                             out, boxes4, area, group, validArr);
    k_zero_mask<<<MASK_TOTAL_WORDS / 256, 256, 0, stream>>>(mask);
    k_tiles<<<(256 * 256) / 8, 256, 0, stream>>>(area, boxes4, group, mask);
    k_nms<<<1, 32, 0, stream>>>(mask, validArr, out);
}